// CombinedLossWithInteractions_34342558499116
// MI455X (gfx1250) — compile-verified
//
#include <hip/hip_runtime.h>
#include <hip/hip_bf16.h>
#include <math.h>

// ---------------------------------------------------------------------------
// CombinedLossWithInteractions for MI455X (gfx1250, wave32).
//   total = 0.6*mse + 0.2*corr + 0.2*bce
//   bce over 8192x8192 logits vs sparse adjacency (262144 edges, set-to-1).
// Strategy:
//   bce_sum = sum(softplus(x)) - sum_{unique edges} x       (exact, bitmap)
//   row stats (Sx,Sy,Sxx,Syy,Sxy) via V_WMMA_F32_16X16X4_F32 with B = ones
//   mse from Sxx - 2Sxy + Syy (node arrays read once).
// ---------------------------------------------------------------------------

static constexpr int       NUM_NODES = 8192;
static constexpr int       FEAT_DIM  = 256;
static constexpr long long NUM_EDGES = 262144;
static constexpr long long NN        = (long long)NUM_NODES * (long long)NUM_NODES; // 67108864

typedef __attribute__((ext_vector_type(2))) float v2f;
typedef __attribute__((ext_vector_type(8))) float v8f;

// acc[0] = bce sum, acc[1] = mse sum, acc[2] = corr (1-r) sum
// ---------------------------------------------------------------------------
// Kernel 1: scatter edge bits (atomicOr de-duplicates, matching .set(1.0))
// ---------------------------------------------------------------------------
__global__ void scatter_edges_kernel(const long long* __restrict__ edges,
                                     unsigned int* __restrict__ bits) {
  long long e = (long long)blockIdx.x * blockDim.x + threadIdx.x;
  if (e < NUM_EDGES) {
    unsigned long long r = (unsigned long long)edges[e];
    unsigned long long c = (unsigned long long)edges[NUM_EDGES + e];
    unsigned long long p = r * (unsigned long long)NUM_NODES + c;
    atomicOr(&bits[p >> 5], 1u << (p & 31u));
  }
}

__device__ __forceinline__ float softplus_stable(float v) {
  // max(v,0) + log1p(exp(-|v|)) : hardware v_exp_f32 / v_log_f32 paths
  return fmaxf(v, 0.0f) + __logf(1.0f + __expf(-fabsf(v)));
}

__device__ __forceinline__ void block_accumulate(float lsum, double* dst) {
  #pragma unroll
  for (int off = 16; off > 0; off >>= 1) lsum += __shfl_xor(lsum, off, 32);
  __shared__ float wsum[8];
  int lane = threadIdx.x & 31, w = threadIdx.x >> 5;
  if (lane == 0) wsum[w] = lsum;
  __syncthreads();
  if (threadIdx.x == 0) {
    float t = 0.0f;
    int nw = blockDim.x >> 5;
    for (int i = 0; i < nw; ++i) t += wsum[i];
    atomicAdd(dst, (double)t);   // global_atomic_add_f64
  }
}

// ---------------------------------------------------------------------------
// Kernel 2: streaming BCE reduction over 256 MiB of logits (bandwidth-bound).
// float4 loads; 4 label bits live in one bitmap word per vector.
// ---------------------------------------------------------------------------
template <bool USE_BITS>
__global__ void bce_kernel(const float* __restrict__ x,
                           const unsigned int* __restrict__ bits,
                           double* __restrict__ acc) {
  const long long nvec = NN >> 2;
  long long tid    = (long long)blockIdx.x * blockDim.x + threadIdx.x;
  long long stride = (long long)gridDim.x * blockDim.x;
  const float4* xv = (const float4*)x;
  float lsum = 0.0f;
  for (long long i = tid; i < nvec; i += stride) {
    float4 v = xv[i];
    __builtin_prefetch(&xv[i + stride], 0, 0);  // global_prefetch_b8
    unsigned int b = 0;
    if (USE_BITS) {
      long long base = i << 2;
      b = bits[base >> 5] >> (base & 31);
    }
    float s = softplus_stable(v.x) + softplus_stable(v.y) +
              softplus_stable(v.z) + softplus_stable(v.w);
    if (USE_BITS) {
      if (b & 1u) s -= v.x;
      if (b & 2u) s -= v.y;
      if (b & 4u) s -= v.z;
      if (b & 8u) s -= v.w;
    }
    lsum += s;
  }
  block_accumulate(lsum, &acc[0]);
}

// Fallback (workspace too small for bitmap): subtract x over edges directly.
// Duplicate edges introduce ~5e-7 relative error in bce.
__global__ void edge_subtract_kernel(const long long* __restrict__ edges,
                                     const float* __restrict__ x,
                                     double* __restrict__ acc) {
  long long e = (long long)blockIdx.x * blockDim.x + threadIdx.x;
  float lsum = 0.0f;
  if (e < NUM_EDGES) {
    long long r = edges[e];
    long long c = edges[NUM_EDGES + e];
    lsum = -x[r * (long long)NUM_NODES + c];
  }
  block_accumulate(lsum, &acc[0]);
}

// ---------------------------------------------------------------------------
// Kernel 3: per-row stats via V_WMMA_F32_16X16X4_F32 with B = ones.
// One wave32 owns a 16-row tile. A layout (16x4 f32): lanes 0-15 hold
// K={k,k+1}, lanes 16-31 hold K={k+2,k+3}, M = lane&15.  D = A*1 + C gives
// the running row sum replicated across all 16 columns, so D VGPR r on
// lanes 0-15 is row r's sum and on lanes 16-31 is row r+8's sum.
// ---------------------------------------------------------------------------
__global__ void row_stats_wmma_kernel(const float* __restrict__ X,
                                      const float* __restrict__ Y,
                                      double* __restrict__ acc) {
  int wave = blockIdx.x * (blockDim.x >> 5) + (threadIdx.x >> 5);
  int lane = threadIdx.x & 31;
  int half = lane >> 4;                 // 0: K=k,k+1  1: K=k+2,k+3
  int row  = wave * 16 + (lane & 15);
  const float* xr = X + (long long)row * FEAT_DIM;
  const float* yr = Y + (long long)row * FEAT_DIM;

  const v2f ones = {1.0f, 1.0f};
  v8f sx = {}, sy = {}, sxx = {}, syy = {}, sxy = {};

  for (int k = 0; k < FEAT_DIM; k += 4) {
    int ka = k + 2 * half;
    v2f ax = *(const v2f*)(xr + ka);    // 8B aligned (ka even, row base 1KB)
    v2f ay = *(const v2f*)(yr + ka);
    v2f axx = ax * ax;
    v2f ayy = ay * ay;
    v2f axy = ax * ay;
    sx  = __builtin_amdgcn_wmma_f32_16x16x4_f32(false, ax,  false, ones, (short)0, sx,  false, false);
    sy  = __builtin_amdgcn_wmma_f32_16x16x4_f32(false, ay,  false, ones, (short)0, sy,  false, false);
    sxx = __builtin_amdgcn_wmma_f32_16x16x4_f32(false, axx, false, ones, (short)0, sxx, false, false);
    syy = __builtin_amdgcn_wmma_f32_16x16x4_f32(false, ayy, false, ones, (short)0, syy, false, false);
    sxy = __builtin_amdgcn_wmma_f32_16x16x4_f32(false, axy, false, ones, (short)0, sxy, false, false);
  }

  // This thread's component r corresponds to row wave*16 + r + 8*half.
  const float invF = 1.0f / (float)FEAT_DIM;
  float local_corr = 0.0f, local_mse = 0.0f;
  #pragma unroll
  for (int r = 0; r < 8; ++r) {
    float Sx = sx[r], Sy = sy[r], Sxx = sxx[r], Syy = syy[r], Sxy = sxy[r];
    local_mse += Sxx - 2.0f * Sxy + Syy;                 // sum (x-y)^2 for row
    float num = Sxy - Sx * Sy * invF;
    float vx  = fmaxf(Sxx - Sx * Sx * invF, 0.0f);
    float vy  = fmaxf(Syy - Sy * Sy * invF, 0.0f);
    float den = sqrtf(vx) * sqrtf(vy) + 1e-12f;
    local_corr += 1.0f - num / den;
  }
  // lanes 0-15 carry rows 0..7 of the tile (identical copies); 16-31 rows 8..15
  float cc = __shfl(local_corr, 0, 32) + __shfl(local_corr, 16, 32);
  float mm = __shfl(local_mse,  0, 32) + __shfl(local_mse,  16, 32);
  if (lane == 0) {
    atomicAdd(&acc[1], (double)mm);
    atomicAdd(&acc[2], (double)cc);
  }
}

// ---------------------------------------------------------------------------
// Kernel 4: combine -> (total, mse, corr, bce)
// ---------------------------------------------------------------------------
__global__ void combine_kernel(const double* __restrict__ acc,
                               float* __restrict__ out) {
  double bce  = acc[0] / (double)NN;
  double mse  = acc[1] / ((double)NUM_NODES * (double)FEAT_DIM);
  double corr = acc[2] / (double)NUM_NODES;
  double total = 0.6 * mse + 0.2 * corr + 0.2 * bce;
  out[0] = (float)total;
  out[1] = (float)mse;
  out[2] = (float)corr;
  out[3] = (float)bce;
}

extern "C" void kernel_launch(void* const* d_in, const int* in_sizes, int n_in,
                              void* d_out, int out_size, void* d_ws, size_t ws_size,
                              hipStream_t stream) {
  const float*     node_pred   = (const float*)d_in[0];
  const float*     node_target = (const float*)d_in[1];
  const float*     inter       = (const float*)d_in[2];
  const long long* edges       = (const long long*)d_in[3];
  float*           out         = (float*)d_out;

  const size_t bits_bytes = (size_t)(NN >> 5) * sizeof(unsigned int); // 8 MiB
  const bool use_bits = ws_size >= bits_bytes + 4 * sizeof(double);

  const int edge_blocks = (int)((NUM_EDGES + 255) / 256);

  if (use_bits) {
    unsigned int* bits = (unsigned int*)d_ws;
    double* acc = (double*)((char*)d_ws + bits_bytes);
    hipMemsetAsync(d_ws, 0, bits_bytes + 4 * sizeof(double), stream);
    scatter_edges_kernel<<<edge_blocks, 256, 0, stream>>>(edges, bits);
    bce_kernel<true><<<2048, 256, 0, stream>>>(inter, bits, acc);
    row_stats_wmma_kernel<<<NUM_NODES / 16 / 8, 256, 0, stream>>>(node_pred, node_target, acc);
    combine_kernel<<<1, 1, 0, stream>>>(acc, out);
  } else {
    double* acc = (double*)d_ws;
    hipMemsetAsync(d_ws, 0, 4 * sizeof(double), stream);
    bce_kernel<false><<<2048, 256, 0, stream>>>(inter, nullptr, acc);
    edge_subtract_kernel<<<edge_blocks, 256, 0, stream>>>(edges, inter, acc);
    row_stats_wmma_kernel<<<NUM_NODES / 16 / 8, 256, 0, stream>>>(node_pred, node_target, acc);
    combine_kernel<<<1, 1, 0, stream>>>(acc, out);
  }
}